// LocalPlanarGuidance_44873818308705
// MI455X (gfx1250) — compile-verified
//
#include <hip/hip_runtime.h>

// Local Planar Guidance upsample, r = 8.
// out[b, 8h+rr, 8w+uu] = x[b,3,h,w] / (x[b,0,h,w]*u(uu) + x[b,1,h,w]*v(rr) + x[b,2,h,w])
// u(k) = v(k) = (k - 3.5) / 8
//
// Memory-bound: ~53 MB moved, ~0.08 GFLOP (intensity ~1.4 FLOP/B) -> store-
// bandwidth limited (~53MB / 23.3TB/s ~ 2.3us floor). Optimize for coalesced
// b128 non-temporal stores; matrix pipes are unreachable by roofline.
//
// Mapping: one block = one input row (b,h); one thread = one 4-float output
// quad across all 8 replicated rows. Every global_store_b128 covers 512
// contiguous bytes per wave32. Input base is block-uniform (SALU + s_clause'd
// global_load_b32). Grid = (Hn, Bn) so b/h come free from hardware (no
// magic-number division on the address critical path).

typedef float float4v __attribute__((ext_vector_type(4)));

namespace {
constexpr int Bn = 16, Cn = 4, Hn = 96, Wn = 128, Rn = 8;
constexpr int HO      = Hn * Rn;          // 768
constexpr int WO      = Wn * Rn;          // 1024
constexpr int PLANE   = Hn * Wn;          // 12288 elements per channel plane
constexpr int IMG_IN  = Cn * PLANE;       // 49152 elements per input image
constexpr int IMG_OUT = HO * WO;          // 786432 elements per output image
constexpr int THREADS = 256;              // = WO/4 column-quads per output row
}

__global__ __launch_bounds__(THREADS) void lpg_upsample_kernel(
    const float* __restrict__ x, float* __restrict__ out)
{
    const int cw   = threadIdx.x;              // column-quad index 0..255
    const int h    = blockIdx.x;               // 0..95
    const int b    = blockIdx.y;               // 0..15
    const int w    = cw >> 1;                  // input column (lane pairs share)
    const int half = cw & 1;                   // which 4 of the 8 u-phases

    // 4 plane values for this input pixel; base address is block-uniform so
    // these lower to scalar-base global_load_b32 in one clause.
    const float* p = x + b * IMG_IN + h * Wn + w;
    __builtin_prefetch(p, 0, 0);               // gfx1250 global_prefetch_b8
    const float x0 = p[0 * PLANE];
    const float x1 = p[1 * PLANE];
    const float x2 = p[2 * PLANE];
    const float x3 = p[3 * PLANE];

    // u(k) = (k - 3.5)/8 for k = 4*half .. 4*half+3 (constants after unroll)
    float u[4];
    #pragma unroll
    for (int j = 0; j < 4; ++j)
        u[j] = (float)(4 * half + j) * 0.125f - 0.4375f;

    float* orow = out + b * IMG_OUT + (h * Rn) * WO + cw * 4;

    #pragma unroll
    for (int rr = 0; rr < Rn; ++rr) {
        const float vv    = (float)rr * 0.125f - 0.4375f;   // v(rr)
        const float basev = fmaf(x1, vv, x2);
        float4v o;
        #pragma unroll
        for (int j = 0; j < 4; ++j) {
            const float d = fmaf(x0, u[j], basev);          // denom > 0.56, well-conditioned
            float r = __builtin_amdgcn_rcpf(d);             // v_rcp_f32 (~1 ulp)
            r = r * fmaf(-d, r, 2.0f);                      // 1 Newton step -> ~fp32 exact
            o[j] = x3 * r;
        }
        // Streaming write-once output: non-temporal b128, 512B contiguous/wave.
        __builtin_nontemporal_store(o, (float4v*)orow);
        orow += WO;
    }
}

extern "C" void kernel_launch(void* const* d_in, const int* in_sizes, int n_in,
                              void* d_out, int out_size, void* d_ws, size_t ws_size,
                              hipStream_t stream) {
    const float* x = (const float*)d_in[0];
    float* out     = (float*)d_out;
    lpg_upsample_kernel<<<dim3(Hn, Bn), THREADS, 0, stream>>>(x, out);
    (void)in_sizes; (void)n_in; (void)out_size; (void)d_ws; (void)ws_size;
}